// VoxelBackBone8x_77558519431892
// MI455X (gfx1250) — compile-verified
//
#include <hip/hip_runtime.h>

typedef float v2f __attribute__((ext_vector_type(2)));
typedef float v8f __attribute__((ext_vector_type(8)));

#define BN_INV 0.9995003746877732f  // float32(1/sqrt(1.001))

// ---------------------------------------------------------------------------
// densify: scatter-add features into dense NDHWC grid, set mask
// ---------------------------------------------------------------------------
__global__ void densify_k(const float* __restrict__ feats, const int* __restrict__ coords,
                          int n, int C, float* __restrict__ x, float* __restrict__ m,
                          int H, int W) {
  long i = (long)blockIdx.x * blockDim.x + threadIdx.x;
  long total = (long)n * C;
  if (i >= total) return;
  int c  = (int)(i % C);
  int pt = (int)(i / C);
  int z = coords[pt * 4 + 1], y = coords[pt * 4 + 2], xx = coords[pt * 4 + 3];
  long vox = ((long)z * H + y) * W + xx;
  atomicAdd(&x[vox * C + c], feats[(long)pt * C + c]);
  if (c == 0) m[vox] = 1.0f;
}

// ---------------------------------------------------------------------------
// mask conv: out = (sum of mask over window) > 0 ? 1 : 0
// ---------------------------------------------------------------------------
__global__ void mask_conv_k(const float* __restrict__ min_, float* __restrict__ mout,
                            int Din, int Hin, int Win, int Dout, int Hout, int Wout,
                            int KD, int KH, int KW, int sd, int sh, int sw,
                            int pd, int ph, int pw) {
  int v = blockIdx.x * blockDim.x + threadIdx.x;
  int M = Dout * Hout * Wout;
  if (v >= M) return;
  int zo = v / (Hout * Wout);
  int rr = v - zo * (Hout * Wout);
  int yo = rr / Wout;
  int xo = rr - yo * Wout;
  float acc = 0.f;
  for (int kd = 0; kd < KD; ++kd) {
    int zi = zo * sd + kd - pd;
    if ((unsigned)zi >= (unsigned)Din) continue;
    for (int kh = 0; kh < KH; ++kh) {
      int yi = yo * sh + kh - ph;
      if ((unsigned)yi >= (unsigned)Hin) continue;
      for (int kw = 0; kw < KW; ++kw) {
        int xi = xo * sw + kw - pw;
        if ((unsigned)xi >= (unsigned)Win) continue;
        acc += min_[((long)zi * Hin + yi) * Win + xi];
      }
    }
  }
  mout[v] = (acc > 0.f) ? 1.f : 0.f;
}

// ---------------------------------------------------------------------------
// residual add with optional slice of b: a(D,Ha,Wa,C) += b[:, :, :Ha, :Wa]
// ---------------------------------------------------------------------------
__global__ void add_sliced_k(float* __restrict__ a, const float* __restrict__ b,
                             int Dd, int Hh, int Ww, int C, int Hb, int Wb) {
  long i = (long)blockIdx.x * blockDim.x + threadIdx.x;
  long M = (long)Dd * Hh * Ww * C;
  if (i >= M) return;
  int c = (int)(i % C);
  long v = i / C;
  int x = (int)(v % Ww);
  long t = v / Ww;
  int y = (int)(t % Hh);
  int z = (int)(t / Hh);
  a[i] += b[(((long)z * Hb + y) * Wb + x) * C + c];
}

__global__ void max_sliced_k(float* __restrict__ a, const float* __restrict__ b,
                             int Dd, int Hh, int Ww, int Hb, int Wb) {
  long i = (long)blockIdx.x * blockDim.x + threadIdx.x;
  long M = (long)Dd * Hh * Ww;
  if (i >= M) return;
  int x = (int)(i % Ww);
  long t = i / Ww;
  int y = (int)(t % Hh);
  int z = (int)(t / Hh);
  float bv = b[((long)z * Hb + y) * Wb + x];
  float av = a[i];
  a[i] = av > bv ? av : bv;
}

// ---------------------------------------------------------------------------
// Implicit-GEMM 3D conv + fused BN + ReLU + mask, fp32 WMMA 16x16x4.
// Templated on kernel dims / Cin / NT: tap decomposition is compile-time (no
// runtime division); the gathered A fragment is loaded once per K-chunk as a
// single 8-byte global_load_b64 and reused across NT 16-wide output-channel
// tiles (NT independent accumulator chains also hide the WMMA->WMMA hazard).
// A frag (16x4 f32): lane<16 holds (M=lane, K={0,1}), lane>=16 (M, K={2,3}).
// B frag (4x16 f32): mirrored. C/D: vgpr r -> M=r (+8 for upper half-wave),
// N = lane&15.
// ---------------------------------------------------------------------------
template <int KD, int KH, int KW, int CIN, int NT>
__global__ __launch_bounds__(256) void conv_wmma_t(
    const float* __restrict__ in, const float* __restrict__ w,
    const float* __restrict__ gam, const float* __restrict__ bet,
    const float* __restrict__ mout, float* __restrict__ out,
    int Din, int Hin, int Win, int Dout, int Hout, int Wout, int Cout,
    int sd, int sh, int sw, int pd, int ph, int pw) {
  const int lane  = threadIdx.x & 31;
  const int wave  = threadIdx.x >> 5;
  const int mrow  = lane & 15;
  const int khalf = lane >> 4;  // 0 -> K pair {0,1}, 1 -> K pair {2,3}
  const int M     = Dout * Hout * Wout;
  const int mtile = blockIdx.x * 8 + wave;
  const int cbase = blockIdx.y * (16 * NT) + mrow;

  int va     = mtile * 16 + mrow;
  bool va_ok = va < M;
  int vc     = va_ok ? va : 0;
  int plane  = Hout * Wout;
  int zo = vc / plane;
  int rr = vc - zo * plane;
  int yo = rr / Wout;
  int xo = rr - yo * Wout;
  int zb = zo * sd - pd, ybs = yo * sh - ph, xb = xo * sw - pw;
  const int base0 = (zb * Hin + ybs) * Win + xb;  // may be negative; guarded

  float sc[NT], bi[NT];
#pragma unroll
  for (int t = 0; t < NT; ++t) {
    sc[t] = gam[cbase + t * 16] * BN_INV;
    bi[t] = bet[cbase + t * 16];
  }

  v8f acc[NT];
#pragma unroll
  for (int t = 0; t < NT; ++t) acc[t] = (v8f){0.f,0.f,0.f,0.f,0.f,0.f,0.f,0.f};

  int kd = 0, kh = 0, kw = 0;
#pragma unroll 1
  for (int tap = 0; tap < KD * KH * KW; ++tap) {
    int zi = zb + kd, yi = ybs + kh, xi = xb + kw;
    bool inb = va_ok && (unsigned)zi < (unsigned)Din &&
               (unsigned)yi < (unsigned)Hin && (unsigned)xi < (unsigned)Win;
    int abase = (base0 + (kd * Hin + kh) * Win + kw) * CIN;  // 32-bit offsets
    const float* __restrict__ wt = w + tap * CIN * Cout;
#pragma unroll
    for (int c0 = 0; c0 < CIN; c0 += 4) {
      int kk = c0 + 2 * khalf;
      // one 8-byte vector load per lane covers the (kk, kk+1) K-pair
      v2f a = inb ? *(const v2f*)(in + abase + kk) : (v2f){0.f, 0.f};
#pragma unroll
      for (int t = 0; t < NT; ++t) {
        v2f b;
        b.x = wt[kk * Cout + cbase + t * 16];
        b.y = wt[(kk + 1) * Cout + cbase + t * 16];
        acc[t] = __builtin_amdgcn_wmma_f32_16x16x4_f32(false, a, false, b,
                                                       (short)0, acc[t],
                                                       false, false);
      }
    }
    if (++kw == KW) { kw = 0; if (++kh == KH) { kh = 0; ++kd; } }
  }

  // fused epilogue: BN * (1/sqrt(1.001)) + beta, ReLU, mask multiply
#pragma unroll
  for (int r = 0; r < 8; ++r) {
    int vm = mtile * 16 + r + 8 * khalf;
    if (vm < M) {
      float mval = mout[vm];
      int ob = vm * Cout + cbase;
#pragma unroll
      for (int t = 0; t < NT; ++t) {
        float y = acc[t][r] * sc[t] + bi[t];
        y = y > 0.f ? y : 0.f;
        out[ob + t * 16] = y * mval;
      }
    }
  }
}

template <int KD, int KH, int KW, int CIN, int NT>
static void conv_launch(const float* in, const float* w, const float* g,
                        const float* b, const float* msk, float* out,
                        int Di, int Hi, int Wi, int Do, int Ho, int Wo, int Co,
                        int sd, int sh, int sw, int pd, int ph, int pw,
                        hipStream_t stream) {
  int M  = Do * Ho * Wo;
  int mt = (M + 15) / 16;
  dim3 grid((mt + 7) / 8, Co / (16 * NT), 1);
  conv_wmma_t<KD, KH, KW, CIN, NT><<<grid, 256, 0, stream>>>(
      in, w, g, b, msk, out, Di, Hi, Wi, Do, Ho, Wo, Co, sd, sh, sw, pd, ph, pw);
}

// ---------------------------------------------------------------------------
// host orchestration
// ---------------------------------------------------------------------------
extern "C" void kernel_launch(void* const* d_in, const int* in_sizes, int n_in,
                              void* d_out, int out_size, void* d_ws, size_t ws_size,
                              hipStream_t stream) {
  (void)n_in; (void)out_size; (void)ws_size;
  const float* vfeat = (const float*)d_in[0];
  const int*   vcrd  = (const int*)d_in[1];
  const float* s0f   = (const float*)d_in[2];
  const int*   s0c   = (const int*)d_in[3];
  const float* s1f   = (const float*)d_in[4];
  const int*   s1c   = (const int*)d_in[5];
  const float* s2f   = (const float*)d_in[6];
  const int*   s2c   = (const int*)d_in[7];
  const float* P[57];
  for (int i = 0; i < 57; ++i) P[i] = (const float*)d_in[8 + i];
  // P: conv_input 0-2, conv_input2 3-5, conv1 6-8,
  //    conv2 9-17, r_conv2 18-26, conv3 27-35, r_conv3 36-44, conv4 45-53, conv_out 54-56

  constexpr int  D0 = 25, H0 = 176, W0 = 176;
  constexpr long NV0 = (long)D0 * H0 * W0;        // 774400
  constexpr int  D1 = 13, H1 = 88, W1 = 88;
  constexpr long NV1 = (long)D1 * H1 * W1;        // 100672
  constexpr int  D2 = 7, H2 = 44, W2 = 44;
  constexpr long NV2 = (long)D2 * H2 * W2;        // 13552
  constexpr int  D2r = 7, H2r = 45, W2r = 45;
  constexpr long NV2r = (long)D2r * H2r * W2r;    // 14175
  constexpr int  D3 = 3, H3 = 22, W3 = 22;
  constexpr long NV3 = (long)D3 * H3 * W3;        // 1452
  constexpr long NV4 = 22L * 22;                  // 484 (D=1)

  float* ws = (float*)d_ws;
  // --- aliased big region (peak = s2 dense, 198 MB) ---
  const long U_SZ = NV0 * 64;
  float* x16  = ws;               // stage A (dead before s1/s2 densify)
  float* s016 = ws + NV0 * 16;
  float* x1v  = ws + NV0 * 32;
  float* s1d  = ws;               // stage B reuse
  float* s2d  = ws;               // stage C reuse
  long off = U_SZ;
  auto alloc = [&](long n) { float* p = ws + off; off += (n + 15) & ~15L; return p; };
  float* x4  = alloc(NV0 * 4);
  float* s04 = alloc(NV0 * 4);
  float* mx  = alloc(NV0);
  float* m0  = alloc(NV0);
  float* m1  = alloc(NV0);
  float* m2f = alloc(NV0);
  float* yb  = alloc(NV1 * 32);
  float* yt  = alloc(NV1 * 32);
  float* rb  = alloc(NV1 * 32);
  float* rt  = alloc(NV1 * 32);
  float* my  = alloc(NV1);
  float* mr  = alloc(NV1);
  float* y3  = alloc(NV2 * 64);
  float* y3t = alloc(NV2 * 64);
  float* my3 = alloc(NV2);
  float* r3  = alloc(NV2r * 64);
  float* r3t = alloc(NV2r * 64);
  float* mr3 = alloc(NV2r);
  float* y4  = alloc(NV3 * 64);
  float* y4t = alloc(NV3 * 64);
  float* my4 = alloc(NV3);
  float* mo  = alloc(NV4);

  auto MASKC = [&](const float* mi, float* mo_, int Di, int Hi, int Wi,
                   int Do, int Ho, int Wo, int KD, int KH, int KW,
                   int sd, int sh, int sw, int pd, int ph, int pw) {
    int M = Do * Ho * Wo;
    mask_conv_k<<<(M + 255) / 256, 256, 0, stream>>>(mi, mo_, Di, Hi, Wi, Do, Ho, Wo,
                                                     KD, KH, KW, sd, sh, sw, pd, ph, pw);
  };
  auto ADD = [&](float* a, const float* b, int Dd, int Hh, int Ww, int C, int Hb, int Wb) {
    long M = (long)Dd * Hh * Ww * C;
    add_sliced_k<<<(int)((M + 255) / 256), 256, 0, stream>>>(a, b, Dd, Hh, Ww, C, Hb, Wb);
  };
  auto MAXM = [&](float* a, const float* b, int Dd, int Hh, int Ww, int Hb, int Wb) {
    long M = (long)Dd * Hh * Ww;
    max_sliced_k<<<(int)((M + 255) / 256), 256, 0, stream>>>(a, b, Dd, Hh, Ww, Hb, Wb);
  };
  auto DENSE = [&](const float* f, const int* c, int n, int C, float* x, float* m) {
    long t = (long)n * C;
    densify_k<<<(int)((t + 255) / 256), 256, 0, stream>>>(f, c, n, C, x, m, H0, W0);
  };

  // ---- stage A: full-res 4ch -> 16ch ----
  hipMemsetAsync(x4, 0, NV0 * 4 * sizeof(float), stream);
  hipMemsetAsync(s04, 0, NV0 * 4 * sizeof(float), stream);
  hipMemsetAsync(mx, 0, NV0 * sizeof(float), stream);
  hipMemsetAsync(m0, 0, NV0 * sizeof(float), stream);
  DENSE(vfeat, vcrd, in_sizes[1] / 4, 4, x4, mx);
  DENSE(s0f, s0c, in_sizes[3] / 4, 4, s04, m0);
  conv_launch<3,3,3,4,1>(x4,  P[0], P[1], P[2], mx, x16,
                         D0,H0,W0, D0,H0,W0,16, 1,1,1, 1,1,1, stream);
  conv_launch<3,3,3,4,1>(s04, P[3], P[4], P[5], m0, s016,
                         D0,H0,W0, D0,H0,W0,16, 1,1,1, 1,1,1, stream);
  conv_launch<3,3,3,16,1>(x16, P[6], P[7], P[8], mx, x1v,
                          D0,H0,W0, D0,H0,W0,16, 1,1,1, 1,1,1, stream);
  ADD(x1v, s016, D0, H0, W0, 16, H0, W0);   // xc1
  MAXM(mx, m0, D0, H0, W0, H0, W0);         // mc1

  // ---- conv2 (full -> half, 16 -> 32) ----
  MASKC(mx, my, D0,H0,W0, D1,H1,W1, 3,3,3, 2,2,2, 1,1,1);
  conv_launch<3,3,3,16,2>(x1v, P[9],  P[10], P[11], my, yb,
                          D0,H0,W0, D1,H1,W1,32, 2,2,2, 1,1,1, stream);
  conv_launch<3,3,3,32,2>(yb,  P[12], P[13], P[14], my, yt,
                          D1,H1,W1, D1,H1,W1,32, 1,1,1, 1,1,1, stream);
  conv_launch<3,3,3,32,2>(yt,  P[15], P[16], P[17], my, yb,
                          D1,H1,W1, D1,H1,W1,32, 1,1,1, 1,1,1, stream);

  // ---- r_conv2 (s1 dense full-res 32ch -> half) ----
  hipMemsetAsync(s1d, 0, NV0 * 32 * sizeof(float), stream);
  hipMemsetAsync(m1, 0, NV0 * sizeof(float), stream);
  DENSE(s1f, s1c, in_sizes[5] / 4, 32, s1d, m1);
  MASKC(m1, mr, D0,H0,W0, D1,H1,W1, 3,3,3, 2,2,2, 1,1,1);
  conv_launch<3,3,3,32,2>(s1d, P[18], P[19], P[20], mr, rb,
                          D0,H0,W0, D1,H1,W1,32, 2,2,2, 1,1,1, stream);
  conv_launch<3,3,3,32,2>(rb,  P[21], P[22], P[23], mr, rt,
                          D1,H1,W1, D1,H1,W1,32, 1,1,1, 1,1,1, stream);
  conv_launch<3,3,3,32,2>(rt,  P[24], P[25], P[26], mr, rb,
                          D1,H1,W1, D1,H1,W1,32, 1,1,1, 1,1,1, stream);
  ADD(yb, rb, D1, H1, W1, 32, H1, W1);      // xc2
  MAXM(my, mr, D1, H1, W1, H1, W1);         // mc2

  // ---- conv3 (half -> quarter, 32 -> 64) ----
  MASKC(my, my3, D1,H1,W1, D2,H2,W2, 3,3,3, 2,2,2, 1,1,1);
  conv_launch<3,3,3,32,4>(yb,  P[27], P[28], P[29], my3, y3,
                          D1,H1,W1, D2,H2,W2,64, 2,2,2, 1,1,1, stream);
  conv_launch<3,3,3,64,4>(y3,  P[30], P[31], P[32], my3, y3t,
                          D2,H2,W2, D2,H2,W2,64, 1,1,1, 1,1,1, stream);
  conv_launch<3,3,3,64,4>(y3t, P[33], P[34], P[35], my3, y3,
                          D2,H2,W2, D2,H2,W2,64, 1,1,1, 1,1,1, stream);

  // ---- r_conv3 (s2 dense full-res 64ch -> (7,45,45), stride 4 pad 2) ----
  hipMemsetAsync(s2d, 0, NV0 * 64 * sizeof(float), stream);
  hipMemsetAsync(m2f, 0, NV0 * sizeof(float), stream);
  DENSE(s2f, s2c, in_sizes[7] / 4, 64, s2d, m2f);
  MASKC(m2f, mr3, D0,H0,W0, D2r,H2r,W2r, 3,3,3, 4,4,4, 2,2,2);
  conv_launch<3,3,3,64,4>(s2d, P[36], P[37], P[38], mr3, r3,
                          D0,H0,W0, D2r,H2r,W2r,64, 4,4,4, 2,2,2, stream);
  conv_launch<3,3,3,64,4>(r3,  P[39], P[40], P[41], mr3, r3t,
                          D2r,H2r,W2r, D2r,H2r,W2r,64, 1,1,1, 1,1,1, stream);
  conv_launch<3,3,3,64,4>(r3t, P[42], P[43], P[44], mr3, r3,
                          D2r,H2r,W2r, D2r,H2r,W2r,64, 1,1,1, 1,1,1, stream);
  ADD(y3, r3, D2, H2, W2, 64, H2r, W2r);    // xc3 (slice [:, :7, :44, :44])
  MAXM(my3, mr3, D2, H2, W2, H2r, W2r);     // mc3

  // ---- conv4 (quarter -> (3,22,22), pad (0,1,1)) ----
  MASKC(my3, my4, D2,H2,W2, D3,H3,W3, 3,3,3, 2,2,2, 0,1,1);
  conv_launch<3,3,3,64,4>(y3,  P[45], P[46], P[47], my4, y4,
                          D2,H2,W2, D3,H3,W3,64, 2,2,2, 0,1,1, stream);
  conv_launch<3,3,3,64,4>(y4,  P[48], P[49], P[50], my4, y4t,
                          D3,H3,W3, D3,H3,W3,64, 1,1,1, 1,1,1, stream);
  conv_launch<3,3,3,64,4>(y4t, P[51], P[52], P[53], my4, y4,
                          D3,H3,W3, D3,H3,W3,64, 1,1,1, 1,1,1, stream);

  // ---- conv_out ((3,22,22) -> (1,22,22), k=(3,1,1), s=(2,1,1), p=0, 64 -> 128) ----
  MASKC(my4, mo, D3,H3,W3, 1,H3,W3, 3,1,1, 2,1,1, 0,0,0);
  conv_launch<3,1,1,64,4>(y4, P[54], P[55], P[56], mo, (float*)d_out,
                          D3,H3,W3, 1,H3,W3,128, 2,1,1, 0,0,0, stream);
}